// TORCHIEKF_66383014527427
// MI455X (gfx1250) — compile-verified
//
#include <hip/hip_runtime.h>
#include <math.h>

// ---------------------------------------------------------------------------
// IEKF sequential scan on one wave (wave32), 21x21 algebra padded to 32x32
// and executed with V_WMMA_F32_16X16X4_F32 on the CDNA5 matrix pipe.
// Round 3: all big matrices live in dynamic LDS (file-scope extern __shared__)
// and the shared noinline mm32 body takes integer offsets, so every access
// keeps addrspace(3) provenance -> clean ds_load/ds_store, no generic-pointer
// null checks / 64-bit address math that polluted round 2.
// ---------------------------------------------------------------------------

#define LD 33                 // row stride (pad: avoid LDS bank conflicts)
#define MS (32 * LD)          // one 32x32 (stride-33) matrix
#define O_P  (0 * MS)
#define O_M1 (1 * MS)
#define O_M2 (2 * MS)
#define O_M3 (3 * MS)
#define O_M4 (4 * MS)
#define O_M5 (5 * MS)
#define O_M6 (6 * MS)
#define LDS_FLOATS (7 * MS)

extern __shared__ float SM[];   // dynamic LDS: 7 padded 32x32 matrices

typedef __attribute__((ext_vector_type(2))) float v2f;
typedef __attribute__((ext_vector_type(8))) float v8f;

__device__ __constant__ float QD[18] = {
  1.0e-3f, 1.0e-3f, 1.0e-3f,      // COV_OMEGA
  1.0e-2f, 1.0e-2f, 1.0e-2f,      // COV_ACC
  6.0e-9f, 6.0e-9f, 6.0e-9f,      // COV_B_OMEGA
  2.0e-4f, 2.0e-4f, 2.0e-4f,      // COV_B_ACC
  1.0e-9f, 1.0e-9f, 1.0e-9f,      // COV_ROT_C_I
  1.0e-9f, 1.0e-9f, 1.0e-9f };    // COV_T_C_I

// ----------------------------- 3x3 helpers ---------------------------------
__device__ __forceinline__ void skew3(float* S, const float* x) {
  S[0] = 0.f;   S[1] = -x[2]; S[2] =  x[1];
  S[3] = x[2];  S[4] = 0.f;   S[5] = -x[0];
  S[6] = -x[1]; S[7] = x[0];  S[8] = 0.f;
}
__device__ __forceinline__ void mat3mul(float* C, const float* A, const float* B) {
  for (int i = 0; i < 3; ++i)
    for (int j = 0; j < 3; ++j) {
      float s = 0.f;
      for (int k = 0; k < 3; ++k) s += A[i*3+k] * B[k*3+j];
      C[i*3+j] = s;
    }
}
__device__ __forceinline__ void mat3Tmul(float* C, const float* A, const float* B) { // A^T B
  for (int i = 0; i < 3; ++i)
    for (int j = 0; j < 3; ++j) {
      float s = 0.f;
      for (int k = 0; k < 3; ++k) s += A[k*3+i] * B[k*3+j];
      C[i*3+j] = s;
    }
}
__device__ __forceinline__ void matvec3(float* y, const float* A, const float* x) {
  for (int i = 0; i < 3; ++i) y[i] = A[i*3+0]*x[0] + A[i*3+1]*x[1] + A[i*3+2]*x[2];
}
__device__ __forceinline__ void matTvec3(float* y, const float* A, const float* x) {
  for (int i = 0; i < 3; ++i) y[i] = A[0*3+i]*x[0] + A[1*3+i]*x[1] + A[2*3+i]*x[2];
}
__device__ __forceinline__ void so3exp3(float* R, const float* phi) {
  float a2 = phi[0]*phi[0] + phi[1]*phi[1] + phi[2]*phi[2];
  float angle = sqrtf(a2);
  float S[9]; skew3(S, phi);
  if (angle < 1e-8f) {
    for (int k = 0; k < 9; ++k) R[k] = S[k];
    R[0] += 1.f; R[4] += 1.f; R[8] += 1.f;
  } else {
    float sa = sinf(angle) / angle;
    float ca = (1.f - cosf(angle)) / a2;
    float SS[9]; mat3mul(SS, S, S);
    for (int k = 0; k < 9; ++k) R[k] = sa * S[k] + ca * SS[k];
    R[0] += 1.f; R[4] += 1.f; R[8] += 1.f;
  }
}
__device__ __forceinline__ void sen3exp_RJ(float* R, float* J, const float* phi) {
  float a2 = phi[0]*phi[0] + phi[1]*phi[1] + phi[2]*phi[2];
  float angle = sqrtf(a2);
  float S[9]; skew3(S, phi);
  if (angle < 1e-8f) {
    for (int k = 0; k < 9; ++k) { R[k] = S[k]; J[k] = 0.5f * S[k]; }
    R[0] += 1.f; R[4] += 1.f; R[8] += 1.f;
    J[0] += 1.f; J[4] += 1.f; J[8] += 1.f;
  } else {
    float sn = sinf(angle), cs = cosf(angle);
    float sa = sn / angle;
    float ca = (1.f - cs) / a2;
    float cj = (angle - sn) / (a2 * angle);
    float SS[9]; mat3mul(SS, S, S);
    for (int k = 0; k < 9; ++k) {
      R[k] = sa * S[k] + ca * SS[k];
      J[k] = ca * S[k] + cj * SS[k];
    }
    R[0] += 1.f; R[4] += 1.f; R[8] += 1.f;
    J[0] += 1.f; J[4] += 1.f; J[8] += 1.f;
  }
}
__device__ __forceinline__ void from_rpy3(float* R, float roll, float pitch, float yaw) {
  float cr = cosf(roll),  sr = sinf(roll);
  float cp = cosf(pitch), sp = sinf(pitch);
  float cy = cosf(yaw),   sy = sinf(yaw);
  float Rz[9] = { cy,-sy,0.f,  sy,cy,0.f,  0.f,0.f,1.f };
  float Ry[9] = { cp,0.f,sp,   0.f,1.f,0.f, -sp,0.f,cp };
  float Rx[9] = { 1.f,0.f,0.f, 0.f,cr,-sr,  0.f,sr,cr };
  float T[9]; mat3mul(T, Rz, Ry); mat3mul(R, T, Rx);
}

// ------------------------- 32x32 WMMA matmul -------------------------------
// SM[cOff] = SM[aOff] @ SM[bOff] for 32x32 (stride LD) matrices held in the
// dynamic-LDS pool; 2x2 tiles of 16x16, K chunks of 4 via
// V_WMMA_F32_16X16X4_F32 (fragment layouts per ISA 7.12.2).
// NOINLINE: single shared body; integer offsets keep LDS addrspace provenance.
__device__ __attribute__((noinline)) void mm32(int cOff, int aOff, int bOff) {
  const int lane = threadIdx.x & 31;
  const int half = lane >> 4;       // 0: lanes 0-15, 1: lanes 16-31
  const int l16  = lane & 15;
#pragma unroll
  for (int mt = 0; mt < 2; ++mt) {
#pragma unroll
    for (int nt = 0; nt < 2; ++nt) {
      v8f acc = {};
#pragma unroll
      for (int kc = 0; kc < 8; ++kc) {
        const int k0 = kc * 4 + 2 * half;     // K pair for this half-wave
        v2f a, b;
        a.x = SM[aOff + (mt*16 + l16)*LD + k0];      // A 16x4: lane=M, vgpr=K
        a.y = SM[aOff + (mt*16 + l16)*LD + k0 + 1];
        b.x = SM[bOff + (k0    )*LD + nt*16 + l16];  // B 4x16: rows by half
        b.y = SM[bOff + (k0 + 1)*LD + nt*16 + l16];
        acc = __builtin_amdgcn_wmma_f32_16x16x4_f32(
            false, a, false, b, (short)0, acc, false, false);
      }
#pragma unroll
      for (int r = 0; r < 8; ++r)                    // C/D: vgpr r -> M=r/r+8
        SM[cOff + (mt*16 + r + 8*half)*LD + nt*16 + l16] = acc[r];
    }
  }
}

// ------------------------- sparse block fills ------------------------------
__device__ __forceinline__ void setF3x3(int bOff, int r, int c, const float* M, float s) {
  for (int i = 0; i < 3; ++i)
    for (int j = 0; j < 3; ++j) SM[bOff + (r+i)*LD + (c+j)] = s * M[i*3+j];
}
__device__ __forceinline__ void setGblk(int aOff, int btOff, int r, int c,
                                        const float* M, float dt) {
  for (int i = 0; i < 3; ++i)
    for (int j = 0; j < 3; ++j) {
      float v = M[i*3+j] * dt;
      SM[aOff  + (r+i)*LD + (c+j)] = v * QD[c+j];  // Gm*dt, cols scaled by Q
      SM[btOff + (c+j)*LD + (r+i)] = v;            // (Gm*dt)^T
    }
}

__device__ __forceinline__ void emit_row(float* out, int n, int lane,
    const float* Rot, const float* vv, const float* pp, const float* bo,
    const float* ba, const float* Rci, const float* tci, float* sRow) {
  if (lane == 0) {
    for (int k = 0; k < 9; ++k) { sRow[k] = Rot[k]; sRow[21+k] = Rci[k]; }
    for (int k = 0; k < 3; ++k) {
      sRow[9+k]  = vv[k]; sRow[12+k] = pp[k]; sRow[15+k] = bo[k];
      sRow[18+k] = ba[k]; sRow[30+k] = tci[k];
    }
  }
  __syncthreads();
  for (int idx = lane; idx < 33; idx += 32) out[(size_t)n*33 + idx] = sRow[idx];
  __syncthreads();
}

// ------------------------------- kernel ------------------------------------
__global__ __launch_bounds__(32, 1) void iekf_scan_kernel(
    const float* __restrict__ t, const float* __restrict__ u,
    const float* __restrict__ mcov, const float* __restrict__ v_mes,
    const float* __restrict__ ang0, const float* __restrict__ w_init,
    float* __restrict__ out, int N) {
  __shared__ float sH[2*24];
  __shared__ float sK[21*2];
  __shared__ float sDx[21];
  __shared__ float sRow[33];

  const int lane = threadIdx.x;
  const float I3[9] = {1.f,0.f,0.f, 0.f,1.f,0.f, 0.f,0.f,1.f};

  // --- replicated scalar state (identical in every lane) ---
  float Rot[9], Rci[9], vv[3], pp[3], bo[3], ba[3], tci[3];
  from_rpy3(Rot, ang0[0], ang0[1], ang0[2]);
  for (int k = 0; k < 9; ++k) Rci[k] = I3[k];
  for (int k = 0; k < 3; ++k) { vv[k] = v_mes[k]; pp[k]=0.f; bo[k]=0.f; ba[k]=0.f; tci[k]=0.f; }

  float beta[6];
  for (int j = 0; j < 6; ++j) beta[j] = powf(10.f, tanhf(w_init[j]));

  // --- P0 ---
  for (int e = lane; e < 32*32; e += 32) { int i = e >> 5, j = e & 31; SM[O_P + i*LD+j] = 0.f; }
  __syncthreads();
  if (lane == 0) {
    SM[O_P + 0*LD+0] = SM[O_P + 1*LD+1] = 1.0e-3f * beta[0];   // COV_ROT0
    SM[O_P + 3*LD+3] = SM[O_P + 4*LD+4] = 1.0e-1f * beta[1];   // COV_V0
    for (int j = 0; j < 3; ++j) {
      SM[O_P + ( 9+j)*LD + ( 9+j)] = 1.0e-1f * beta[2];        // COV_B_OMEGA0
      SM[O_P + (12+j)*LD + (12+j)] = 1.0e-1f * beta[3];        // COV_B_ACC0
      SM[O_P + (15+j)*LD + (15+j)] = 1.0e-6f * beta[4];        // COV_ROT_C_I0
      SM[O_P + (18+j)*LD + (18+j)] = 1.0e-5f * beta[5];        // COV_T_C_I0
    }
  }
  __syncthreads();

  emit_row(out, 0, lane, Rot, vv, pp, bo, ba, Rci, tci, sRow);

  for (int n = 1; n < N; ++n) {
    const float dt  = t[n] - t[n-1];
    const float* un = u + (size_t)n * 6;
    float ui[6];
    for (int j = 0; j < 6; ++j) ui[j] = un[j];
    const float mc0 = mcov[(size_t)n*2 + 0];
    const float mc1 = mcov[(size_t)n*2 + 1];

    // ---------------- propagation scalars (replicated) ----------------
    float am[3]  = { ui[3]-ba[0], ui[4]-ba[1], ui[5]-ba[2] };
    float acc[3]; matvec3(acc, Rot, am); acc[2] += -9.80665f;
    float v_new[3], p_new[3];
    for (int j = 0; j < 3; ++j) {
      v_new[j] = vv[j] + acc[j]*dt;
      p_new[j] = pp[j] + vv[j]*dt + 0.5f*acc[j]*dt*dt;
    }
    float omg[3] = { (ui[0]-bo[0])*dt, (ui[1]-bo[1])*dt, (ui[2]-bo[2])*dt };
    float dRm[9]; so3exp3(dRm, omg);
    float Rot_new[9]; mat3mul(Rot_new, Rot, dRm);
    float sv[9], sp_[9]; skew3(sv, vv); skew3(sp_, pp);
    float vsr[9], psr[9]; mat3mul(vsr, sv, Rot); mat3mul(psr, sp_, Rot);

    // ---------------- GQG^T = (Gm dt Q)(Gm dt)^T (one WMMA matmul) -----
    for (int e = lane; e < 32*32; e += 32) {
      int i = e >> 5, j = e & 31;
      SM[O_M4 + i*LD+j] = 0.f; SM[O_M5 + i*LD+j] = 0.f;
    }
    __syncthreads();
    if (lane == 0) {
      setGblk(O_M4, O_M5, 0, 0, Rot, dt);
      setGblk(O_M4, O_M5, 3, 3, Rot, dt);
      setGblk(O_M4, O_M5, 3, 0, vsr, dt);
      setGblk(O_M4, O_M5, 6, 0, psr, dt);
      setGblk(O_M4, O_M5,  9,  6, I3, dt);
      setGblk(O_M4, O_M5, 12,  9, I3, dt);
      setGblk(O_M4, O_M5, 15, 12, I3, dt);
      setGblk(O_M4, O_M5, 18, 15, I3, dt);
    }
    __syncthreads();
    mm32(O_M6, O_M4, O_M5);           // sM6 = G Q G^T (21x21 valid)
    __syncthreads();

    // ---------------- F*dt, F2, F3 -------------------------------------
    for (int e = lane; e < 32*32; e += 32) { int i=e>>5, j=e&31; SM[O_M1 + i*LD+j]=0.f; }
    __syncthreads();
    if (lane == 0) {
      float gv[3] = {0.f, 0.f, -9.80665f};
      float sg[9]; skew3(sg, gv);
      setF3x3(O_M1, 3, 0, sg, dt);
      for (int j = 0; j < 3; ++j) SM[O_M1 + (6+j)*LD + (3+j)] = dt;  // I3*dt
      setF3x3(O_M1, 3, 12, Rot, -dt);
      setF3x3(O_M1, 0,  9, Rot, -dt);
      setF3x3(O_M1, 3,  9, vsr, -dt);
      setF3x3(O_M1, 6,  9, psr, -dt);
    }
    __syncthreads();
    mm32(O_M2, O_M1, O_M1);           // F^2
    __syncthreads();
    mm32(O_M3, O_M2, O_M1);           // F^3
    __syncthreads();

    // Phi = I + F + F2/2 + F3/6 (M4), PhiT (M5), Pn = GQG + P (M6)
    for (int e = lane; e < 32*32; e += 32) {
      int i = e >> 5, j = e & 31;
      float phi = ((i == j) && (i < 21) ? 1.f : 0.f)
                + SM[O_M1 + i*LD+j] + 0.5f*SM[O_M2 + i*LD+j]
                + (1.f/6.f)*SM[O_M3 + i*LD+j];
      SM[O_M4 + i*LD+j] = phi;
      SM[O_M5 + j*LD+i] = phi;
      SM[O_M6 + i*LD+j] += SM[O_P + i*LD+j];
    }
    __syncthreads();
    mm32(O_M1, O_M4, O_M6);           // Phi @ Pn
    __syncthreads();
    mm32(O_P, O_M1, O_M5);            // (Phi @ Pn) @ Phi^T
    __syncthreads();

    // ---------------- measurement update -------------------------------
    float Rot_body[9]; mat3mul(Rot_body, Rot_new, Rci);
    float v_imu[3];    matTvec3(v_imu, Rot_new, v_new);
    float om[3] = { ui[0]-bo[0], ui[1]-bo[1], ui[2]-bo[2] };
    float stc[9]; skew3(stc, tci);
    float tb[3];  matTvec3(tb, Rci, v_imu);
    float skom[3]; matvec3(skom, stc, om);
    float v_body1 = tb[1] + skom[1];
    float v_body2 = tb[2] + skom[2];
    float Om[9];  skew3(Om, om);
    float svi[9]; skew3(svi, v_imu);
    float Hvi[9]; mat3Tmul(Hvi, Rci, svi);    // Rci^T @ skew(v_imu)
    const float r0 = -v_body1, r1 = -v_body2;

    for (int e = lane; e < 32*32; e += 32) { int i=e>>5, j=e&31; SM[O_M5 + i*LD+j]=0.f; }
    __syncthreads();
    if (lane == 0) {
      for (int k = 0; k < 24; ++k) { sH[k] = 0.f; sH[24+k] = 0.f; }
      for (int a = 0; a < 2; ++a)
        for (int c = 0; c < 3; ++c) {
          sH[a*24 + 3  + c] =  Rot_body[c*3 + (1+a)];  // Rot_body^T rows 1,2
          sH[a*24 + 15 + c] =  Hvi[(1+a)*3 + c];
          sH[a*24 + 9  + c] =  stc[(1+a)*3 + c];
          sH[a*24 + 18 + c] = -Om[(1+a)*3 + c];
        }
      for (int k = 0; k < 21; ++k) {                    // H^T into M5
        SM[O_M5 + k*LD + 0] = sH[k];
        SM[O_M5 + k*LD + 1] = sH[24 + k];
      }
    }
    __syncthreads();
    mm32(O_M6, O_P, O_M5);             // P @ H^T  (cols 0,1 valid)
    __syncthreads();

    // S = H P H^T + Rm (2x2, replicated), K = PHt S^{-1}, dx = K r
    float S00 = mc0, S01 = 0.f, S10 = 0.f, S11 = mc1;
    for (int k = 0; k < 21; ++k) {
      float h0 = sH[k], h1 = sH[24+k];
      float q0 = SM[O_M6 + k*LD+0], q1 = SM[O_M6 + k*LD+1];
      S00 += h0*q0; S01 += h0*q1; S10 += h1*q0; S11 += h1*q1;
    }
    float det = S00*S11 - S01*S10;
    float i00 =  S11/det, i01 = -S01/det, i10 = -S10/det, i11 = S00/det;
    if (lane < 21) {
      float q0 = SM[O_M6 + lane*LD+0], q1 = SM[O_M6 + lane*LD+1];
      float k0 = q0*i00 + q1*i10;
      float k1 = q0*i01 + q1*i11;
      sK[lane*2+0] = k0; sK[lane*2+1] = k1;
      sDx[lane] = k0*r0 + k1*r1;
    }
    __syncthreads();

    // I - K H into M4, transpose into M5
    for (int e = lane; e < 32*32; e += 32) {
      int i = e >> 5, j = e & 31;
      float val = 0.f;
      if (i < 21 && j < 21)
        val = ((i == j) ? 1.f : 0.f) - sK[i*2]*sH[j] - sK[i*2+1]*sH[24+j];
      SM[O_M4 + i*LD+j] = val;
      SM[O_M5 + j*LD+i] = val;
    }
    __syncthreads();
    mm32(O_M1, O_M4, O_P);             // (I-KH) @ P
    __syncthreads();
    mm32(O_M6, O_M1, O_M5);            // ... @ (I-KH)^T
    __syncthreads();
    for (int e = lane; e < 32*32; e += 32) {           // + K Rm K^T
      int i = e >> 5, j = e & 31;
      if (i < 21 && j < 21)
        SM[O_M6 + i*LD+j] += sK[i*2]*mc0*sK[j*2] + sK[i*2+1]*mc1*sK[j*2+1];
    }
    __syncthreads();
    for (int e = lane; e < 32*32; e += 32) {           // symmetrize into P
      int i = e >> 5, j = e & 31;
      SM[O_P + i*LD+j] = (i < 21 && j < 21)
                       ? 0.5f*(SM[O_M6 + i*LD+j] + SM[O_M6 + j*LD+i]) : 0.f;
    }
    __syncthreads();

    // ---------------- state retraction (replicated) --------------------
    float dx[21];
#pragma unroll
    for (int k = 0; k < 21; ++k) dx[k] = sDx[k];
    float phiv[3] = { dx[0], dx[1], dx[2] };
    float dR[9], Jm[9]; sen3exp_RJ(dR, Jm, phiv);
    float xv[3] = { dx[3], dx[4], dx[5] };
    float xp[3] = { dx[6], dx[7], dx[8] };
    float c0[3], c1[3]; matvec3(c0, Jm, xv); matvec3(c1, Jm, xp);
    float Rt[9]; mat3mul(Rt, dR, Rot_new);
    for (int k = 0; k < 9; ++k) Rot[k] = Rt[k];
    float tv[3]; matvec3(tv, dR, v_new);
    float tp[3]; matvec3(tp, dR, p_new);
    for (int k = 0; k < 3; ++k) {
      vv[k] = tv[k] + c0[k];
      pp[k] = tp[k] + c1[k];
      bo[k] += dx[9+k];
      ba[k] += dx[12+k];
      tci[k] += dx[18+k];
    }
    float phic[3] = { dx[15], dx[16], dx[17] };
    float dRc[9]; so3exp3(dRc, phic);
    float Rc[9]; mat3mul(Rc, dRc, Rci);
    for (int k = 0; k < 9; ++k) Rci[k] = Rc[k];

    emit_row(out, n, lane, Rot, vv, pp, bo, ba, Rci, tci, sRow);
  }
}

extern "C" void kernel_launch(void* const* d_in, const int* in_sizes, int n_in,
                              void* d_out, int out_size, void* d_ws, size_t ws_size,
                              hipStream_t stream) {
  const float* t      = (const float*)d_in[0];
  const float* u      = (const float*)d_in[1];
  const float* mcov   = (const float*)d_in[2];
  const float* v_mes  = (const float*)d_in[3];
  // d_in[4] = p_mes (unused by reference _run)
  const float* ang0   = (const float*)d_in[5];
  const float* w_init = (const float*)d_in[6];
  const int N = in_sizes[0];
  const size_t lds_bytes = (size_t)LDS_FLOATS * sizeof(float);
  iekf_scan_kernel<<<dim3(1), dim3(32), lds_bytes, stream>>>(
      t, u, mcov, v_mes, ang0, w_init, (float*)d_out, N);
}